// SOM_46454366273643
// MI455X (gfx1250) — compile-verified
//
#include <hip/hip_runtime.h>
#include <stdint.h>

typedef __attribute__((ext_vector_type(2))) float v2f;
typedef __attribute__((ext_vector_type(4))) float v4f;
typedef __attribute__((ext_vector_type(8))) float v8f;

#define SOM_DIM   512
#define SOM_NROW  65536     // 256*256 neurons
#define SOM_ALPHA 0.3f
#define SOM_SIGMA 128.0f

// ---------------------------------------------------------------------------
// Pass 0: reset the packed (d2bits<<32 | row) argmin slot in workspace.
// ---------------------------------------------------------------------------
__global__ void som_init_ws(unsigned long long* ws) {
    if (threadIdx.x == 0) ws[0] = 0xFFFFFFFFFFFFFFFFULL;
}

// ---------------------------------------------------------------------------
// Pass 1: BMU search. One wave32 owns 16 codebook rows (M). The w.x dot
// products come from V_WMMA_F32_16X16X4_F32 (A = 16x4 weight tile, B = x
// replicated across all 16 N columns); ||w||^2 accumulates on the VALU from
// the same loaded float2s. Rank rows by ||w||^2 - 2*dot (||x||^2 constant).
// ---------------------------------------------------------------------------
__global__ __launch_bounds__(256) void som_bmu_wmma(
        const float* __restrict__ x,
        const float* __restrict__ w,
        unsigned long long* __restrict__ ws) {
    const int lane = threadIdx.x & 31;
    const int wave = threadIdx.x >> 5;
    const int half = lane >> 4;          // K half: lanes 0-15 -> K%4 in {0,1}, 16-31 -> {2,3}
    const int r    = lane & 15;          // local row (M) for the A operand
    const int rowbase = (blockIdx.x * 8 + wave) * 16;
    const float* __restrict__ wr = w + (size_t)(rowbase + r) * SOM_DIM;

    v8f c0 = {};                         // two accumulators to break the WMMA chain
    v8f c1 = {};
    float wn2 = 0.0f;                    // this lane's half of ||w_row||^2

#pragma unroll 2
    for (int k = 0; k < SOM_DIM; k += 8) {
        const int kp0 = k + 2 * half;
        const int kp1 = k + 4 + 2 * half;
        v2f a0 = *(const v2f*)(wr + kp0);    // A tile: VGPR0=K even, VGPR1=K odd
        v2f b0 = *(const v2f*)(x  + kp0);    // B tile: x broadcast over N (hot in L0)
        v2f a1 = *(const v2f*)(wr + kp1);
        v2f b1 = *(const v2f*)(x  + kp1);
        wn2 = fmaf(a0.x, a0.x, wn2);
        wn2 = fmaf(a0.y, a0.y, wn2);
        wn2 = fmaf(a1.x, a1.x, wn2);
        wn2 = fmaf(a1.y, a1.y, wn2);
        // 8 args: (neg_a, A, neg_b, B, c_mod, C, reuse_a, reuse_b); f32 WMMA
        // only supports C-negate, so the -2x is applied after the K loop.
        c0 = __builtin_amdgcn_wmma_f32_16x16x4_f32(false, a0, false, b0,
                                                   (short)0, c0, false, false);
        c1 = __builtin_amdgcn_wmma_f32_16x16x4_f32(false, a1, false, b1,
                                                   (short)0, c1, false, false);
    }
    v8f c = c0 + c1;

    // Combine the two K-halves of ||w||^2: lane L now holds row rowbase+(L&15).
    wn2 += __shfl_xor(wn2, 16);

    // Each lane scores the 8 rows of its half (D layout: VGPR j, lanes<16 ->
    // M=j, lanes>=16 -> M=j+8; all N columns identical). Redundant across
    // lanes, but the wave covers all 16 rows.
    unsigned long long best = 0xFFFFFFFFFFFFFFFFULL;
#pragma unroll
    for (int j = 0; j < 8; ++j) {
        const int m = j + 8 * half;                  // local row index
        const float wnm = __shfl(wn2, m);            // ||w_row||^2 lives at lane m
        const float dot = c[j];
        const float v = fmaf(-2.0f, dot, wnm);       // = d2 - ||x||^2 (same order)
        unsigned u = __float_as_uint(v);
        u ^= (unsigned)((int)u >> 31) | 0x80000000u; // monotone float->uint map
        const unsigned long long key =
            ((unsigned long long)u << 32) | (unsigned)(rowbase + m);
        best = key < best ? key : best;
    }
    // wave32 min-reduce; ties pick the smallest row index (matches jnp.argmin)
#pragma unroll
    for (int off = 16; off > 0; off >>= 1) {
        const unsigned long long o = __shfl_xor(best, off);
        best = o < best ? o : best;
    }
    if (lane == 0) atomicMin(ws, best);
}

// ---------------------------------------------------------------------------
// Pass 2: streaming Gaussian-neighborhood update, one float4 per thread.
// W re-read mostly hits the 192MB L2 (pass 1 warmed all 134MB of it).
// ---------------------------------------------------------------------------
__global__ __launch_bounds__(256) void som_update(
        const float* __restrict__ x,
        const float* __restrict__ w,
        float* __restrict__ out,
        const unsigned long long* __restrict__ ws) {
    const unsigned bmu = (unsigned)(ws[0] & 0xFFFFFFFFu);  // wave-uniform scalar load
    const float bi = (float)(bmu >> 8);
    const float bj = (float)(bmu & 255u);

    const size_t gid = (size_t)blockIdx.x * blockDim.x + threadIdx.x;
    const int neuron = (int)(gid >> 7);      // 512/4 = 128 float4 per neuron
    const int c4     = (int)(gid & 127);

    const float rate    = 1.0f - (float)neuron * (1.0f / 65536.0f);
    const float alpha_t = SOM_ALPHA * rate;
    const float sigma_t = SOM_SIGMA * rate;
    const float di = bi - (float)(neuron >> 8);
    const float dj = bj - (float)(neuron & 255);
    const float ld2 = di * di + dj * dj;
    const float s = alpha_t * __expf(-ld2 / (2.0f * sigma_t * sigma_t));

    const v4f wv = ((const v4f*)w)[gid];
    const v4f xv = ((const v4f*)x)[c4];      // 2KB vector, hot in L0
    v4f ov;
    ov.x = fmaf(s, xv.x - wv.x, wv.x);
    ov.y = fmaf(s, xv.y - wv.y, wv.y);
    ov.z = fmaf(s, xv.z - wv.z, wv.z);
    ov.w = fmaf(s, xv.w - wv.w, wv.w);
    ((v4f*)out)[gid] = ov;
}

// ---------------------------------------------------------------------------
extern "C" void kernel_launch(void* const* d_in, const int* in_sizes, int n_in,
                              void* d_out, int out_size, void* d_ws, size_t ws_size,
                              hipStream_t stream) {
    (void)in_sizes; (void)n_in; (void)out_size; (void)ws_size;
    const float* x = (const float*)d_in[0];   // [512] f32
    const float* w = (const float*)d_in[1];   // [65536, 512] f32
    float* out = (float*)d_out;               // [65536, 512] f32
    unsigned long long* ws = (unsigned long long*)d_ws;

    som_init_ws<<<1, 1, 0, stream>>>(ws);
    // 65536 rows / (8 waves * 16 rows) = 512 blocks
    som_bmu_wmma<<<512, 256, 0, stream>>>(x, w, ws);
    // 65536 * 128 float4 / 256 threads = 32768 blocks
    som_update<<<32768, 256, 0, stream>>>(x, w, out, ws);
}